// GCN_Block_4243427688839
// MI455X (gfx1250) — compile-verified
//
#include <hip/hip_runtime.h>
#include <math.h>

typedef float v2f __attribute__((ext_vector_type(2)));
typedef float v8f __attribute__((ext_vector_type(8)));

#define B 8
#define C 64
#define N 4096
#define BN_EPS 1e-5f

// ---------------------------------------------------------------------------
// Kernel 1: per-batch scalar  s_b = sum_n relu(tanh(w))^2 ; alpha/beta
// ---------------------------------------------------------------------------
__global__ void batch_scalar_kernel(const float* __restrict__ w, float* __restrict__ ab) {
    __shared__ float red[256];
    const int b = blockIdx.x;
    float s = 0.f;
    for (int n = threadIdx.x; n < N; n += 256) {
        float v = tanhf(w[b * N + n]);
        v = fmaxf(v, 0.f);
        s += v * v;
    }
    red[threadIdx.x] = s;
    __syncthreads();
    for (int off = 128; off > 0; off >>= 1) {
        if (threadIdx.x < off) red[threadIdx.x] += red[threadIdx.x + off];
        __syncthreads();
    }
    if (threadIdx.x == 0) {
        float sb = red[0];
        float d  = (float)N * sb + 1.0f;
        ab[b]     = sb / d;    // alpha_b
        ab[8 + b] = 1.0f / d;  // beta_b
    }
}

// ---------------------------------------------------------------------------
// Kernel 2: colsum[b,c] = sum_n x[b,c,n]
// ---------------------------------------------------------------------------
__global__ void colsum_kernel(const float* __restrict__ x, float* __restrict__ cs) {
    __shared__ float red[256];
    const int c = blockIdx.x;
    const int b = blockIdx.y;
    const float* xp = x + ((size_t)b * C + c) * N;
    float s = 0.f;
    for (int n = threadIdx.x; n < N; n += 256) s += xp[n];
    red[threadIdx.x] = s;
    __syncthreads();
    for (int off = 128; off > 0; off >>= 1) {
        if (threadIdx.x < off) red[threadIdx.x] += red[threadIdx.x + off];
        __syncthreads();
    }
    if (threadIdx.x == 0) cs[b * C + c] = red[0];
}

// ---------------------------------------------------------------------------
// Kernel 3: t[b,o] = alpha_b * sum_c W[o,c]*cs[b,c] + conv_b[o]
// ---------------------------------------------------------------------------
__global__ void tmat_kernel(const float* __restrict__ conv_w, const float* __restrict__ conv_b,
                            const float* __restrict__ ab, const float* __restrict__ cs,
                            float* __restrict__ t) {
    const int b = blockIdx.x;
    const int o = threadIdx.x;  // 64 threads
    float acc = 0.f;
    #pragma unroll 8
    for (int c = 0; c < C; ++c) acc += conv_w[o * C + c] * cs[b * C + c];
    t[b * C + o] = ab[b] * acc + conv_b[o];
}

// ---------------------------------------------------------------------------
// Kernel 4 (WMMA main): y[b,o,n] = beta_b * sum_c W[o,c] * x[b,c,n] + t[b,o]
// grid = (N/32, B), block = 256 (8 waves). Wave (mt,nt): 16x16 output tile,
// rows mt*16.., cols nbase+nt*16.. ; K=64 via 16x V_WMMA_F32_16X16X4_F32.
// y is written into d_out (reused as scratch for the BN pass).
// ---------------------------------------------------------------------------
__global__ void __launch_bounds__(256)
wmma_main_kernel(const float* __restrict__ x, const float* __restrict__ conv_w,
                 const float* __restrict__ ab, const float* __restrict__ t,
                 float* __restrict__ y) {
    const int b     = blockIdx.y;
    const int nbase = blockIdx.x * 32;
    const int wave  = threadIdx.x >> 5;
    const int lane  = threadIdx.x & 31;
    const int mt    = wave & 3;            // M tile 0..3  (rows mt*16..)
    const int nt    = wave >> 2;           // N subtile 0..1
    const int hi    = lane >> 4;           // lane half
    const int lm    = lane & 15;
    const int col   = nbase + nt * 16 + lm;
    const int m     = mt * 16 + lm;        // A-matrix row (output channel)

    const float beta = ab[8 + b];
    const float* __restrict__ Xb = x + (size_t)b * C * N;

    v8f acc = {};
    #pragma unroll
    for (int kk = 0; kk < 16; ++kk) {
        const int k = kk * 4 + 2 * hi;     // ISA 16x4 f32 A layout: lanes 0-15 K={0,1}, 16-31 K={2,3}
        v2f a, bm;
        a.x  = conv_w[m * C + k];
        a.y  = conv_w[m * C + k + 1];
        bm.x = Xb[(size_t)k * N + col];    // B 4x16 layout mirrors A: row K in lane-half, N across lanes
        bm.y = Xb[(size_t)(k + 1) * N + col];
        acc = __builtin_amdgcn_wmma_f32_16x16x4_f32(
            /*neg_a=*/false, a, /*neg_b=*/false, bm,
            /*c_mod=*/(short)0, acc, /*reuse_a=*/false, /*reuse_b=*/false);
    }

    float* __restrict__ Yb = y + (size_t)b * C * N;
    #pragma unroll
    for (int r = 0; r < 8; ++r) {
        const int row = mt * 16 + r + 8 * hi;   // C/D layout: VGPR r -> M=r (lanes 0-15), M=r+8 (16-31)
        Yb[(size_t)row * N + col] = beta * acc[r] + t[b * C + row];
    }
}

// ---------------------------------------------------------------------------
// Kernel 5: BN stats per channel over (B, N):  mean[o], rstd[o]
// ---------------------------------------------------------------------------
__global__ void bnstats_kernel(const float* __restrict__ y, float* __restrict__ mr) {
    __shared__ float rs[256], rq[256];
    const int o = blockIdx.x;
    float s = 0.f, q = 0.f;
    for (int idx = threadIdx.x; idx < B * N; idx += 256) {
        const int bb = idx >> 12;          // idx / N
        const int nn = idx & (N - 1);
        float v = y[((size_t)bb * C + o) * N + nn];
        s += v;
        q += v * v;
    }
    rs[threadIdx.x] = s;
    rq[threadIdx.x] = q;
    __syncthreads();
    for (int off = 128; off > 0; off >>= 1) {
        if (threadIdx.x < off) {
            rs[threadIdx.x] += rs[threadIdx.x + off];
            rq[threadIdx.x] += rq[threadIdx.x + off];
        }
        __syncthreads();
    }
    if (threadIdx.x == 0) {
        const float inv = 1.0f / (float)(B * N);
        float mean = rs[0] * inv;
        float var  = rq[0] * inv - mean * mean;
        mr[o]      = mean;
        mr[C + o]  = rsqrtf(var + BN_EPS);
    }
}

// ---------------------------------------------------------------------------
// Kernel 6: out = relu((y - mean) * rstd * gamma + beta)  (in place on d_out)
// ---------------------------------------------------------------------------
__global__ void finalize_kernel(float* __restrict__ y, const float* __restrict__ mr,
                                const float* __restrict__ gamma, const float* __restrict__ beta) {
    const size_t i4 = ((size_t)blockIdx.x * blockDim.x + threadIdx.x) * 4;
    const int o = (int)((i4 >> 12) & (C - 1));   // channel of these 4 consecutive elements
    const float mean = mr[o];
    const float scl  = mr[C + o] * gamma[o];
    const float sft  = beta[o];
    float4 v = *(const float4*)(y + i4);
    v.x = fmaxf((v.x - mean) * scl + sft, 0.f);
    v.y = fmaxf((v.y - mean) * scl + sft, 0.f);
    v.z = fmaxf((v.z - mean) * scl + sft, 0.f);
    v.w = fmaxf((v.w - mean) * scl + sft, 0.f);
    *(float4*)(y + i4) = v;
}

// ---------------------------------------------------------------------------
extern "C" void kernel_launch(void* const* d_in, const int* in_sizes, int n_in,
                              void* d_out, int out_size, void* d_ws, size_t ws_size,
                              hipStream_t stream) {
    const float* x      = (const float*)d_in[0];  // [B,C,N,1]
    const float* w      = (const float*)d_in[1];  // [B,N]
    const float* conv_w = (const float*)d_in[2];  // [C,C,1,1]
    const float* conv_b = (const float*)d_in[3];  // [C]
    const float* gamma  = (const float*)d_in[4];  // [C]
    const float* bbeta  = (const float*)d_in[5];  // [C]
    float* out = (float*)d_out;                   // [B,C,N,1] — also used as y scratch
    float* wsf = (float*)d_ws;
    float* ab = wsf;          // alpha[8], beta[8]
    float* cs = wsf + 16;     // colsum [B,C]
    float* t  = wsf + 528;    // t [B,C]
    float* mr = wsf + 1040;   // mean[C], rstd[C]

    batch_scalar_kernel<<<B, 256, 0, stream>>>(w, ab);
    colsum_kernel<<<dim3(C, B), 256, 0, stream>>>(x, cs);
    tmat_kernel<<<B, C, 0, stream>>>(conv_w, conv_b, ab, cs, t);
    wmma_main_kernel<<<dim3(N / 32, B), 256, 0, stream>>>(x, conv_w, ab, t, out);
    bnstats_kernel<<<C, 256, 0, stream>>>(out, mr);
    finalize_kernel<<<(B * C * N) / (256 * 4), 256, 0, stream>>>(out, mr, gamma, bbeta);
}